// GAT_85383949845215
// MI455X (gfx1250) — compile-verified
//
#include <hip/hip_runtime.h>
#include <math.h>

// ---------------- problem constants (match reference) ----------------
#define F1     128   // F_IN
#define HD     256   // HEADS * HID
#define NHEADS 8
#define HID    32
#define NC     40    // num classes (layer-2 out dim, heads=1)
#define SLOPE  0.2f

typedef __attribute__((ext_vector_type(2))) float    v2f;
typedef __attribute__((ext_vector_type(8))) float    v8f;
typedef __attribute__((ext_vector_type(4))) unsigned v4u;
typedef __attribute__((ext_vector_type(4))) int      v4i;
typedef __attribute__((ext_vector_type(8))) int      v8i;

#if defined(__has_builtin)
#if __has_builtin(__builtin_amdgcn_tensor_load_to_lds)
#define GAT_HAVE_TDM 1
#endif
#endif

// Monotonic float<->u32 encoding so float max == unsigned-int max.
__device__ __forceinline__ unsigned encf(float f) {
  unsigned u = __float_as_uint(f);
  return (u & 0x80000000u) ? ~u : (u | 0x80000000u);
}
__device__ __forceinline__ float decf(unsigned u) {
  return (u & 0x80000000u) ? __uint_as_float(u & 0x7fffffffu)
                           : __uint_as_float(~u);
}
__device__ __forceinline__ float lrelu(float x) { return x > 0.f ? x : SLOPE * x; }

// edge slot -> (src, dst); slots >= E are the appended self-loops
__device__ __forceinline__ void edge_sd(const int* __restrict__ EI, int E, int e,
                                        int& s, int& d) {
  if (e < E) { s = EI[e]; d = EI[E + e]; } else { s = d = e - E; }
}

// Opaque LDS read: ds_load_b64 at a raw byte address. The compiler cannot see
// (and therefore cannot fold away) the TDM's writes into LDS, so the A-operand
// reads must be opaque too. Wait folded in since the compiler doesn't track
// the DScnt of an asm load.
__device__ __forceinline__ v2f lds_read_b64(unsigned byte_addr) {
  v2f r;
  asm volatile("ds_load_b64 %0, %1\n\ts_wait_dscnt 0x0"
               : "=v"(r) : "v"(byte_addr));
  return r;
}

#ifdef GAT_HAVE_TDM
// TDM: DMA a [16 x width] f32 tile (row stride == width elements) from global
// into LDS at lds_base, inserting 4 pad DWORDs after every `width` DWORDs so
// the LDS image has row stride width+4 (bank-conflict-free: (4*row+col)%64).
// pad_code: D# pad_interval encoding (0=2,1=4,...,6=128,7=256 DWORDs).
// Rows past `rows_left` are OOB in dim1 and read as zero by the TDM.
__device__ __forceinline__ void tdm_load_tile(const float* gsrc,
                                              unsigned lds_base,
                                              unsigned rows_left,
                                              unsigned width,
                                              unsigned pad_code) {
  unsigned long long ga = (unsigned long long)(size_t)gsrc;
  v4u g0;
  g0[0] = 1u;                                   // count=1, user descriptor
  g0[1] = lds_base;                             // lds_addr (bytes)
  g0[2] = (unsigned)(ga & 0xFFFFFFFFu);         // global_addr[31:0]
  g0[3] = (unsigned)((ga >> 32) & 0x01FFFFFFu)  // global_addr[56:32]
          | (2u << 30);                         // type = 2 ("image")
  v8i g1;
  g1[0] = (int)((2u << 16)          // data_size = 4 bytes
                | (1u << 20)        // pad_enable
                | (pad_code << 22)  // pad_interval
                | (3u << 25));      // pad_amount = 4 DWORDs
  g1[1] = (int)((width & 0xFFFFu) << 16);                       // tensor_dim0 lo16
  g1[2] = (int)((width >> 16) | ((rows_left & 0xFFFFu) << 16)); // dim0 hi | dim1 lo
  g1[3] = (int)((rows_left >> 16) | (width << 16));             // dim1 hi | tile_dim0
  g1[4] = 16;                                                   // tile_dim1 = 16
  g1[5] = (int)width;                                           // tensor_dim0_stride
  g1[6] = 0;
  g1[7] = 0;
  v4i z4 = {0, 0, 0, 0};
#if __clang_major__ >= 23
  v8i z8 = {0, 0, 0, 0, 0, 0, 0, 0};
  __builtin_amdgcn_tensor_load_to_lds(g0, g1, z4, z4, z8, 0);
#else
  __builtin_amdgcn_tensor_load_to_lds(g0, g1, z4, z4, 0);
#endif
#if __has_builtin(__builtin_amdgcn_s_wait_tensorcnt)
  __builtin_amdgcn_s_wait_tensorcnt(0);
#else
  asm volatile("s_wait_tensorcnt 0x0" ::: "memory");
#endif
}
#endif  // GAT_HAVE_TDM

// ---------------- utility ----------------
__global__ void GAT_zero(float* __restrict__ p, size_t n) {
  size_t i = (size_t)blockIdx.x * blockDim.x + threadIdx.x;
  size_t st = (size_t)gridDim.x * blockDim.x;
  for (; i < n; i += st) p[i] = 0.f;
}

// ---------------- GEMM1: H1[N,256] = X[N,128] @ W1[128,256] (f32 WMMA) -------
// block = (32,4): 4 waves share one 16-row A tile in LDS (TDM-loaded), each
// wave owns a 16-col tile of W1. grid = (ceil(N/16), 4) -> covers 256 cols.
__global__ __launch_bounds__(128) void GAT_gemm1(const float* __restrict__ X,
                                                 const float* __restrict__ W,
                                                 float* __restrict__ H, int N) {
  __shared__ float sA[16 * 132];  // row stride 132: (4*row+col)%64 conflict-free
  const int row0 = blockIdx.x * 16;
  // Low 32 bits of a generic pointer into the LDS aperture == DS byte offset
  // (ISA 10.2). Also forces sA to stay allocated & escaped.
  const unsigned sbase = (unsigned)(size_t)&sA[0];
#ifdef GAT_HAVE_TDM
  if (threadIdx.y == 0)
    tdm_load_tile(X + (size_t)row0 * F1, sbase, (unsigned)(N - row0), F1,
                  6u /*pad every 128 DW*/);
#else
  const int tid = threadIdx.y * 32 + threadIdx.x;
  for (int idx = tid; idx < 16 * (F1 / 4); idx += 128) {
    int r = idx / (F1 / 4), c4 = idx % (F1 / 4);
    int row = row0 + r;
    if (row >= N) row = N - 1;
    float4 v = ((const float4*)(X + (size_t)row * F1))[c4];
    float* dst = &sA[r * 132 + c4 * 4];
    dst[0] = v.x; dst[1] = v.y; dst[2] = v.z; dst[3] = v.w;
  }
#endif
  __syncthreads();

  const int lane = threadIdx.x;
  const int l15 = lane & 15;
  const int kp = (lane >> 4) * 2;  // lanes 0-15: K=0,1  lanes 16-31: K=2,3
  const int col0 = (blockIdx.y * 4 + threadIdx.y) * 16;
  const float* __restrict__ wcol = W + col0 + l15;
  const unsigned arow = sbase + (unsigned)(l15 * 132 + kp) * 4u;

  v8f acc = {};
#pragma unroll
  for (int k = 0; k < F1; k += 4) {
    v2f a = lds_read_b64(arow + (unsigned)k * 4u);  // A[m][k+kp], A[m][k+kp+1]
    v2f b;
    b.x = wcol[(size_t)(k + kp) * HD];
    b.y = wcol[(size_t)(k + kp + 1) * HD];
    acc = __builtin_amdgcn_wmma_f32_16x16x4_f32(false, a, false, b, (short)0,
                                                acc, false, false);
  }
  const int mhi = (lane >> 4) * 8;
#pragma unroll
  for (int r = 0; r < 8; ++r) {
    int row = row0 + r + mhi;
    if (row < N) H[(size_t)row * HD + col0 + l15] = acc[r];
  }
}

// ---------------- GEMM2: HG[N,40] = A[N,256] @ W2[256,40] (f32 WMMA) ---------
// block = (32,3): 3 waves share one 16-row A tile (TDM-loaded); col tiles
// 0/16/32, cols >= 40 masked (zero B, skip store). EXEC full through WMMA.
__global__ __launch_bounds__(96) void GAT_gemm2(const float* __restrict__ A,
                                                const float* __restrict__ W,
                                                float* __restrict__ HG, int N) {
  __shared__ float sA[16 * 260];
  const int row0 = blockIdx.x * 16;
  const unsigned sbase = (unsigned)(size_t)&sA[0];
#ifdef GAT_HAVE_TDM
  if (threadIdx.y == 0)
    tdm_load_tile(A + (size_t)row0 * HD, sbase, (unsigned)(N - row0), HD,
                  7u /*pad every 256 DW*/);
#else
  const int tid = threadIdx.y * 32 + threadIdx.x;
  for (int idx = tid; idx < 16 * (HD / 4); idx += 96) {
    int r = idx / (HD / 4), c4 = idx % (HD / 4);
    int row = row0 + r;
    if (row >= N) row = N - 1;
    float4 v = ((const float4*)(A + (size_t)row * HD))[c4];
    float* dst = &sA[r * 260 + c4 * 4];
    dst[0] = v.x; dst[1] = v.y; dst[2] = v.z; dst[3] = v.w;
  }
#endif
  __syncthreads();

  const int lane = threadIdx.x;
  const int l15 = lane & 15;
  const int kp = (lane >> 4) * 2;
  const int col = threadIdx.y * 16 + l15;
  const bool cok = col < NC;
  const unsigned arow = sbase + (unsigned)(l15 * 260 + kp) * 4u;

  v8f acc = {};
#pragma unroll 8
  for (int k = 0; k < HD; k += 4) {
    v2f a = lds_read_b64(arow + (unsigned)k * 4u);
    v2f b;
    b.x = cok ? W[(size_t)(k + kp) * NC + col] : 0.f;
    b.y = cok ? W[(size_t)(k + kp + 1) * NC + col] : 0.f;
    acc = __builtin_amdgcn_wmma_f32_16x16x4_f32(false, a, false, b, (short)0,
                                                acc, false, false);
  }
  if (cok) {
    const int mhi = (lane >> 4) * 8;
#pragma unroll
    for (int r = 0; r < 8; ++r) {
      int row = row0 + r + mhi;
      if (row < N) HG[(size_t)row * NC + col] = acc[r];
    }
  }
}

// ---------------- attention logits ----------------
__global__ void GAT_alpha1(const float* __restrict__ H1,
                           const float* __restrict__ a_src,
                           const float* __restrict__ a_dst,
                           float* __restrict__ as, float* __restrict__ ad, int N) {
  int t = blockIdx.x * blockDim.x + threadIdx.x;
  if (t >= N * NHEADS) return;
  int node = t >> 3, h = t & 7;
  const float* hp = H1 + (size_t)node * HD + h * HID;
  const float* s = a_src + h * HID;
  const float* d = a_dst + h * HID;
  float vs = 0.f, vd = 0.f;
#pragma unroll
  for (int i = 0; i < HID; ++i) { float x = hp[i]; vs += x * s[i]; vd += x * d[i]; }
  as[t] = vs; ad[t] = vd;
}

__global__ void GAT_alpha2(const float* __restrict__ HG,
                           const float* __restrict__ a_src,
                           const float* __restrict__ a_dst,
                           float* __restrict__ as, float* __restrict__ ad, int N) {
  int t = blockIdx.x * blockDim.x + threadIdx.x;
  if (t >= N) return;
  const float* hp = HG + (size_t)t * NC;
  float vs = 0.f, vd = 0.f;
#pragma unroll
  for (int i = 0; i < NC; ++i) { float x = hp[i]; vs += x * a_src[i]; vd += x * a_dst[i]; }
  as[t] = vs; ad[t] = vd;
}

// ---------------- layer-1 edge passes (8 heads) ----------------
__global__ void GAT_max1(const int* __restrict__ EI, int E, int N,
                         const float* __restrict__ as, const float* __restrict__ ad,
                         unsigned* __restrict__ m) {
  long long t = (long long)blockIdx.x * blockDim.x + threadIdx.x;
  if (t >= (long long)(E + N) * NHEADS) return;
  int e = (int)(t >> 3), h = (int)(t & 7), s, d;
  edge_sd(EI, E, e, s, d);
  float ev = lrelu(as[(size_t)s * NHEADS + h] + ad[(size_t)d * NHEADS + h]);
  atomicMax(&m[(size_t)d * NHEADS + h], encf(ev));
}

__global__ void GAT_sum1(const int* __restrict__ EI, int E, int N,
                         const float* __restrict__ as, const float* __restrict__ ad,
                         const unsigned* __restrict__ m, float* __restrict__ ssum) {
  long long t = (long long)blockIdx.x * blockDim.x + threadIdx.x;
  if (t >= (long long)(E + N) * NHEADS) return;
  int e = (int)(t >> 3), h = (int)(t & 7), s, d;
  edge_sd(EI, E, e, s, d);
  float ev = lrelu(as[(size_t)s * NHEADS + h] + ad[(size_t)d * NHEADS + h]);
  atomicAdd(&ssum[(size_t)d * NHEADS + h], __expf(ev - decf(m[(size_t)d * NHEADS + h])));
}

__global__ void GAT_agg1(const int* __restrict__ EI, int E, int N,
                         const float* __restrict__ as, const float* __restrict__ ad,
                         const unsigned* __restrict__ m, const float* __restrict__ ssum,
                         const float* __restrict__ H1, float* __restrict__ AG) {
  long long t = (long long)blockIdx.x * blockDim.x + threadIdx.x;
  if (t >= (long long)(E + N) * NHEADS) return;
  int e = (int)(t >> 3), h = (int)(t & 7), s, d;
  edge_sd(EI, E, e, s, d);
  size_t di = (size_t)d * NHEADS + h;
  float ev = lrelu(as[(size_t)s * NHEADS + h] + ad[di]);
  float alpha = __expf(ev - decf(m[di])) / (ssum[di] + 1e-16f);
  const float* hs = H1 + (size_t)s * HD + h * HID;
  float* o = AG + (size_t)d * HD + h * HID;
#pragma unroll
  for (int i = 0; i < HID; ++i) atomicAdd(&o[i], hs[i] * alpha);
}

// in-place h2 = elu(agg1 + b1)
__global__ void GAT_elu(float* __restrict__ AG, const float* __restrict__ b, size_t n) {
  size_t i = (size_t)blockIdx.x * blockDim.x + threadIdx.x;
  if (i >= n) return;
  float v = AG[i] + b[i & (HD - 1)];
  AG[i] = v > 0.f ? v : (__expf(v) - 1.f);
}

// ---------------- layer-2 edge passes (1 head) ----------------
__global__ void GAT_max2(const int* __restrict__ EI, int E, int N,
                         const float* __restrict__ as, const float* __restrict__ ad,
                         unsigned* __restrict__ m) {
  int t = blockIdx.x * blockDim.x + threadIdx.x;
  if (t >= E + N) return;
  int s, d;
  edge_sd(EI, E, t, s, d);
  atomicMax(&m[d], encf(lrelu(as[s] + ad[d])));
}

__global__ void GAT_sum2(const int* __restrict__ EI, int E, int N,
                         const float* __restrict__ as, const float* __restrict__ ad,
                         const unsigned* __restrict__ m, float* __restrict__ ssum) {
  int t = blockIdx.x * blockDim.x + threadIdx.x;
  if (t >= E + N) return;
  int s, d;
  edge_sd(EI, E, t, s, d);
  float ev = lrelu(as[s] + ad[d]);
  atomicAdd(&ssum[d], __expf(ev - decf(m[d])));
}

__global__ void GAT_agg2(const int* __restrict__ EI, int E, int N,
                         const float* __restrict__ as, const float* __restrict__ ad,
                         const unsigned* __restrict__ m, const float* __restrict__ ssum,
                         const float* __restrict__ HG, float* __restrict__ OUT) {
  long long t = (long long)blockIdx.x * blockDim.x + threadIdx.x;
  if (t >= (long long)(E + N) * 5) return;   // 5 chunks of 8 cols = 40
  int e = (int)(t / 5), ch = (int)(t % 5);
  int s, d;
  edge_sd(EI, E, e, s, d);
  float ev = lrelu(as[s] + ad[d]);
  float alpha = __expf(ev - decf(m[d])) / (ssum[d] + 1e-16f);
  const float* hs = HG + (size_t)s * NC + ch * 8;
  float* o = OUT + (size_t)d * NC + ch * 8;
#pragma unroll
  for (int i = 0; i < 8; ++i) atomicAdd(&o[i], hs[i] * alpha);
}

__global__ void GAT_bias2(float* __restrict__ OUT, const float* __restrict__ b, size_t n) {
  size_t i = (size_t)blockIdx.x * blockDim.x + threadIdx.x;
  if (i >= n) return;
  OUT[i] += b[i % NC];
}

// ---------------- host launcher ----------------
extern "C" void kernel_launch(void* const* d_in, const int* in_sizes, int n_in,
                              void* d_out, int out_size, void* d_ws, size_t ws_size,
                              hipStream_t stream) {
  const float* X   = (const float*)d_in[0];
  const int*   EI  = (const int*)d_in[1];   // edge_index [2,E], int32
  const float* W1  = (const float*)d_in[2];
  const float* A1s = (const float*)d_in[3];
  const float* A1d = (const float*)d_in[4];
  const float* B1  = (const float*)d_in[5];
  const float* W2  = (const float*)d_in[6];
  const float* A2s = (const float*)d_in[7];
  const float* A2d = (const float*)d_in[8];
  const float* B2  = (const float*)d_in[9];
  float* OUT = (float*)d_out;

  const int N = in_sizes[0] / F1;
  const int E = in_sizes[1] / 2;
  (void)n_in; (void)out_size; (void)ws_size;

  // carve scratch
  size_t off = 0;
  char* w = (char*)d_ws;
  auto carve = [&](size_t bytes) -> void* {
    void* p = w + off;
    off = (off + bytes + 255) & ~(size_t)255;
    return p;
  };
  float*    H1  = (float*)carve((size_t)N * HD * 4);
  float*    AG1 = (float*)carve((size_t)N * HD * 4);
  float*    as1 = (float*)carve((size_t)N * NHEADS * 4);
  float*    ad1 = (float*)carve((size_t)N * NHEADS * 4);
  unsigned* m1  = (unsigned*)carve((size_t)N * NHEADS * 4);
  float*    s1  = (float*)carve((size_t)N * NHEADS * 4);
  float*    HG  = (float*)carve((size_t)N * NC * 4);
  float*    as2 = (float*)carve((size_t)N * 4);
  float*    ad2 = (float*)carve((size_t)N * 4);
  unsigned* m2  = (unsigned*)carve((size_t)N * 4);
  float*    s2  = (float*)carve((size_t)N * 4);

  const int TB = 256;
  auto blocks = [&](long long n) { return (unsigned)((n + TB - 1) / TB); };

  // zero the atomic accumulators (harness poisons ws/out with 0xAA)
  GAT_zero<<<blocks((long long)N * HD), TB, 0, stream>>>(AG1, (size_t)N * HD);
  GAT_zero<<<blocks((long long)N * NHEADS), TB, 0, stream>>>((float*)m1, (size_t)N * NHEADS);
  GAT_zero<<<blocks((long long)N * NHEADS), TB, 0, stream>>>(s1, (size_t)N * NHEADS);
  GAT_zero<<<blocks(N), TB, 0, stream>>>((float*)m2, (size_t)N);
  GAT_zero<<<blocks(N), TB, 0, stream>>>(s2, (size_t)N);
  GAT_zero<<<blocks((long long)N * NC), TB, 0, stream>>>(OUT, (size_t)N * NC);

  // layer 1
  dim3 gg1((N + 15) / 16, 4), bg1(32, 4);
  GAT_gemm1<<<gg1, bg1, 0, stream>>>(X, W1, H1, N);
  GAT_alpha1<<<blocks((long long)N * NHEADS), TB, 0, stream>>>(H1, A1s, A1d, as1, ad1, N);
  long long t1 = (long long)(E + N) * NHEADS;
  GAT_max1<<<blocks(t1), TB, 0, stream>>>(EI, E, N, as1, ad1, m1);
  GAT_sum1<<<blocks(t1), TB, 0, stream>>>(EI, E, N, as1, ad1, m1, s1);
  GAT_agg1<<<blocks(t1), TB, 0, stream>>>(EI, E, N, as1, ad1, m1, s1, H1, AG1);
  GAT_elu<<<blocks((long long)N * HD), TB, 0, stream>>>(AG1, B1, (size_t)N * HD);

  // layer 2
  dim3 gg2((N + 15) / 16, 1), bg2(32, 3);
  GAT_gemm2<<<gg2, bg2, 0, stream>>>(AG1, W2, HG, N);
  GAT_alpha2<<<blocks(N), TB, 0, stream>>>(HG, A2s, A2d, as2, ad2, N);
  GAT_max2<<<blocks(E + N), TB, 0, stream>>>(EI, E, N, as2, ad2, m2);
  GAT_sum2<<<blocks(E + N), TB, 0, stream>>>(EI, E, N, as2, ad2, m2, s2);
  GAT_agg2<<<blocks((long long)(E + N) * 5), TB, 0, stream>>>(EI, E, N, as2, ad2, m2, s2, HG, OUT);
  GAT_bias2<<<blocks((long long)N * NC), TB, 0, stream>>>(OUT, B2, (size_t)N * NC);
}